// FoutNet_39006892982902
// MI455X (gfx1250) — compile-verified
//
#include <hip/hip_runtime.h>
#include <stdint.h>

// out = x @ (Wc + Wn) + b   — the scatter-mean in the reference is dead code.
// GEMM M=100000, N=128, K=128 in fp32 via V_WMMA_F32_16X16X4_F32 (wave32).
// B fragments staged to LDS per block via the Tensor Data Mover (TDM).

typedef __attribute__((ext_vector_type(2))) float        v2f;
typedef __attribute__((ext_vector_type(8))) float        v8f;
typedef __attribute__((ext_vector_type(4))) unsigned int v4u;
typedef __attribute__((ext_vector_type(8))) int          v8i;
typedef __attribute__((ext_vector_type(4))) int          v4i;

#define DIM     128
#define NTILES  8                           // 128/16 column tiles per wave
#define KSTEPS  32                          // 128/4  k-steps
#define WAVESZ  32
#define WFRAG_N (KSTEPS * NTILES * WAVESZ)  // 8192 float2 = 64 KB

// Toolchain hedge: therock headers ship amd_gfx1250_TDM.h and a 6-arg builtin;
// ROCm 7.2 has the 5-arg form.
#if __has_include(<hip/amd_detail/amd_gfx1250_TDM.h>)
#define TDM_LOAD_TO_LDS(g0, g1, g2, g3) \
    __builtin_amdgcn_tensor_load_to_lds((g0), (g1), (g2), (g3), (v8i){0,0,0,0,0,0,0,0}, 0)
#else
#define TDM_LOAD_TO_LDS(g0, g1, g2, g3) \
    __builtin_amdgcn_tensor_load_to_lds((g0), (g1), (g2), (g3), 0)
#endif

// ---------------------------------------------------------------------------
// Kernel 1: build W = Wc + Wn directly in per-lane B-fragment order for
// V_WMMA_F32_16X16X4_F32.  B is 4x16 (KxN), 2 VGPRs:
//   lane L: N = L%16, holds K = koff, koff+1 with koff = 2*(L/16).
// wfrag[(kstep*NTILES + ntile)*32 + lane] = { W[k][n], W[k+1][n] }   (64 KB)
// ---------------------------------------------------------------------------
__global__ __launch_bounds__(256) void pack_w_frag(const float* __restrict__ Wc,
                                                   const float* __restrict__ Wn,
                                                   float2* __restrict__ wfrag) {
    int t = blockIdx.x * blockDim.x + threadIdx.x;    // 0 .. 8191
    if (t >= WFRAG_N) return;
    int lane  = t & 31;
    int ntile = (t >> 5) & (NTILES - 1);
    int kstep = t >> 8;
    int n = ntile * 16 + (lane & 15);
    int k = kstep * 4 + ((lane >> 4) << 1);           // koff = 0 or 2
    float lo = Wc[(size_t)k * DIM + n]       + Wn[(size_t)k * DIM + n];
    float hi = Wc[(size_t)(k + 1) * DIM + n] + Wn[(size_t)(k + 1) * DIM + n];
    wfrag[t] = make_float2(lo, hi);
}

// ---------------------------------------------------------------------------
// Kernel 2: tiled GEMM + bias.
//   block = 256 threads = 8 wave32; each wave: 16 rows x 128 cols of output.
//   Stage 64 KB fragment buffer global->LDS with one TDM descriptor (wave 0),
//   s_wait_tensorcnt 0, barrier; inner loop: x via VMEM, B via DS pipe.
// ---------------------------------------------------------------------------
__global__ __launch_bounds__(256) void gemm_bias_wmma(const float*  __restrict__ x,
                                                      const float2* __restrict__ wfrag,
                                                      const float*  __restrict__ bias,
                                                      float* __restrict__ out,
                                                      int M) {
    __shared__ float2 lds_wf[WFRAG_N];                // 64 KB

    const int lane = threadIdx.x & 31;
    const int wave = threadIdx.x >> 5;

    if (wave == 0) {
        // Tensor DMA descriptor (ISA 8.3/8.4): 1-D tile viewed as 2-D
        // tensor 16384x1 of 4-byte elements, tile = whole tensor.
        const uint64_t ga  = (uint64_t)(uintptr_t)wfrag;
        const uint32_t lds = (uint32_t)(uintptr_t)&lds_wf[0];   // low 32b = LDS offset

        v4u g0;
        g0.x = 1u;                                    // count=1, user descriptor
        g0.y = lds;                                   // lds_addr (bytes)
        g0.z = (uint32_t)ga;                          // global_addr[31:0]
        g0.w = (uint32_t)((ga >> 32) & 0x01FFFFFFu)   // global_addr[56:32]
             | (2u << 30);                            // type = 2 ("image")

        v8i g1;
        g1[0] = (int)(2u << 16);                      // wg_mask=0, data_size=2 (4B)
        g1[1] = (int)(16384u << 16);                  // tensor_dim0[15:0] = 16384
        g1[2] = (int)(1u << 16);                      // dim0 hi=0, tensor_dim1 = 1
        g1[3] = (int)(16384u << 16);                  // dim1 hi=0, tile_dim0 = 16384
        g1[4] = 1;                                    // tile_dim1 = 1, tile_dim2 = 0
        g1[5] = 16384;                                // tensor_dim0_stride[31:0]
        g1[6] = 0;                                    // stride0 hi, stride1 lo
        g1[7] = 0;

        const v4i gz = (v4i){0, 0, 0, 0};             // groups 2/3 unused (2-D)
        TDM_LOAD_TO_LDS(g0, g1, gz, gz);
        __builtin_amdgcn_s_wait_tensorcnt(0);         // TENSORcnt == 0
    }
    __syncthreads();                                  // LDS tile visible to all waves

    const int row_base = blockIdx.x * 128 + wave * 16;   // wave-uniform
    if (row_base < M) {                               // uniform branch: EXEC all-1s
        const int m    = row_base + (lane & 15);      // this lane's A row
        const int koff = (lane >> 4) << 1;            // 0 or 2

        v8f acc[NTILES];
#pragma unroll
        for (int n = 0; n < NTILES; ++n) acc[n] = (v8f){};

        const float* __restrict__ xrow = x + (size_t)m * DIM + koff;
        __builtin_prefetch(xrow, 0, 0);               // global_prefetch_b8
        __builtin_prefetch(xrow + 64, 0, 0);

        const float2* lw = &lds_wf[lane];

#pragma unroll 4
        for (int ks = 0; ks < KSTEPS; ++ks) {
            const float2 af = *(const float2*)(xrow + ks * 4);
            v2f a; a.x = af.x; a.y = af.y;
#pragma unroll
            for (int n = 0; n < NTILES; ++n) {
                const float2 bf = lw[(ks * NTILES + n) * WAVESZ];  // ds_load_b64
                v2f b; b.x = bf.x; b.y = bf.y;
                // (neg_a, A, neg_b, B, c_mod, C, reuse_a, reuse_b)
                acc[n] = __builtin_amdgcn_wmma_f32_16x16x4_f32(
                    false, a, false, b, (short)0, acc[n], false, false);
            }
        }

        // Epilogue: D layout (VGPR r -> row r + 8*(lane/16), col = lane%16),
        // coalesced b32 stores, add bias.
        const int ncol  = lane & 15;
        const int rhalf = (lane >> 4) << 3;
#pragma unroll
        for (int n = 0; n < NTILES; ++n) {
            const float bv = bias[n * 16 + ncol];
#pragma unroll
            for (int r = 0; r < 8; ++r) {
                const int mrow = row_base + rhalf + r;
                out[(size_t)mrow * DIM + n * 16 + ncol] = acc[n][r] + bv;
            }
        }
    }
}

// ---------------------------------------------------------------------------
// Inputs (setup_inputs order): x[100000*128] f32, edge_index[2*640000] i64
// (dead code), Wc[128*128] f32, Wn[128*128] f32, b[128] f32.
// Output: [100000*128] f32.
// ---------------------------------------------------------------------------
extern "C" void kernel_launch(void* const* d_in, const int* in_sizes, int n_in,
                              void* d_out, int out_size, void* d_ws, size_t ws_size,
                              hipStream_t stream) {
    const float* x  = (const float*)d_in[0];
    // d_in[1] = edge_index (int64) — unused: gamma is dead code in the reference
    const float* Wc = (const float*)d_in[2];
    const float* Wn = (const float*)d_in[3];
    const float* b  = (const float*)d_in[4];
    float* out      = (float*)d_out;
    float2* wfrag   = (float2*)d_ws;                  // 64 KB fragment buffer

    const int M = in_sizes[0] / DIM;                  // 100000

    pack_w_frag<<<(WFRAG_N + 255) / 256, 256, 0, stream>>>(Wc, Wn, wfrag);

    const int grid = (M + 127) / 128;                 // 782 blocks of 8 waves
    gemm_bias_wmma<<<grid, 256, 0, stream>>>(x, wfrag, b, out, M);
}